// MoEBlock_3633542333083
// MI455X (gfx1250) — compile-verified
//
#include <hip/hip_runtime.h>
#include <math.h>

// Problem constants (B=2, S=2048 -> T=4096)
#define TOK  4096
#define DIM  1024
#define FF   4096
#define NEXP 8
#define CAP  2048   // C = 2*K*T/E

typedef float v2f __attribute__((ext_vector_type(2)));
typedef float v8f __attribute__((ext_vector_type(8)));
typedef int   v4i __attribute__((ext_vector_type(4)));

#if defined(__has_builtin)
#if __has_builtin(__builtin_amdgcn_global_load_async_to_lds_b128) && \
    __has_builtin(__builtin_amdgcn_global_load_async_to_lds_b32)
#define USE_ASYNC_LDS 1
#endif
#endif

typedef __attribute__((address_space(1))) v4i* gas128_t;
typedef __attribute__((address_space(1))) int* gas32_t;
typedef __attribute__((address_space(3))) v4i* las128_t;
typedef __attribute__((address_space(3))) int* las32_t;

__device__ __forceinline__ void wait_async0() {
#if __has_builtin(__builtin_amdgcn_s_wait_asynccnt)
    __builtin_amdgcn_s_wait_asynccnt(0);
#else
    asm volatile("s_wait_asynccnt 0x0" ::: "memory");
#endif
}

// ---------------------------------------------------------------------------
// 1) Gating: logits = x @ Wg, softmax, top-2, normalized weights.
//    One wave32 per token.
// ---------------------------------------------------------------------------
__global__ __launch_bounds__(256) void gate_kernel(
    const float* __restrict__ x, const float* __restrict__ Wg,
    int* __restrict__ idx1, int* __restrict__ idx2,
    float* __restrict__ w1, float* __restrict__ w2)
{
    const int lane = threadIdx.x & 31;
    const int t = blockIdx.x * 8 + (threadIdx.x >> 5);
    if (t >= TOK) return;

    float p[NEXP];
#pragma unroll
    for (int e = 0; e < NEXP; ++e) p[e] = 0.0f;

    const float* xr = x + (size_t)t * DIM;
    for (int j = lane; j < DIM; j += 32) {
        const float xv = xr[j];
        const float4 g0 = *(const float4*)(Wg + (size_t)j * NEXP);
        const float4 g1 = *(const float4*)(Wg + (size_t)j * NEXP + 4);
        p[0] += xv * g0.x; p[1] += xv * g0.y; p[2] += xv * g0.z; p[3] += xv * g0.w;
        p[4] += xv * g1.x; p[5] += xv * g1.y; p[6] += xv * g1.z; p[7] += xv * g1.w;
    }
#pragma unroll
    for (int off = 16; off >= 1; off >>= 1) {
#pragma unroll
        for (int e = 0; e < NEXP; ++e) p[e] += __shfl_xor(p[e], off);
    }

    if (lane == 0) {
        float mx = p[0];
#pragma unroll
        for (int e = 1; e < NEXP; ++e) mx = fmaxf(mx, p[e]);
        float g[NEXP]; float s = 0.0f;
#pragma unroll
        for (int e = 0; e < NEXP; ++e) { g[e] = expf(p[e] - mx); s += g[e]; }
        const float inv = 1.0f / s;

        int i1 = 0; float best = p[0];
#pragma unroll
        for (int e = 1; e < NEXP; ++e) if (p[e] > best) { best = p[e]; i1 = e; }
        int i2 = (i1 == 0) ? 1 : 0; float best2 = p[i2];
#pragma unroll
        for (int e = 0; e < NEXP; ++e)
            if (e != i1 && p[e] > best2) { best2 = p[e]; i2 = e; }

        const float g1v = g[i1] * inv;
        const float g2v = g[i2] * inv;
        const float denom = g1v + g2v + 1e-9f;
        idx1[t] = i1; idx2[t] = i2;
        w1[t] = g1v / denom; w2[t] = g2v / denom;
    }
}

// ---------------------------------------------------------------------------
// 2) GShard slot assignment: token-order prefix counts per expert.
// ---------------------------------------------------------------------------
__global__ __launch_bounds__(256) void scan_kernel(
    const int* __restrict__ idx1, const int* __restrict__ idx2,
    int* __restrict__ pos1, int* __restrict__ pos2)
{
    const int e = blockIdx.x;
    __shared__ int waveCnt[8];
    __shared__ int runningBase;
    const int tid = threadIdx.x;
    const int lane = tid & 31;
    const int wave = tid >> 5;

    if (tid == 0) runningBase = 0;
    __syncthreads();

    for (int pass = 0; pass < 2; ++pass) {
        const int* idxA = pass ? idx2 : idx1;
        int* posA = pass ? pos2 : pos1;
        for (int c0 = 0; c0 < TOK; c0 += 256) {
            const int t = c0 + tid;
            const bool m = (idxA[t] == e);
            const unsigned mask = (unsigned)__ballot(m);
            const int inWave = __builtin_popcount(mask & ((1u << lane) - 1u));
            if (lane == 0) waveCnt[wave] = __builtin_popcount(mask);
            __syncthreads();
            int prefix = 0, blockTotal = 0;
#pragma unroll
            for (int wv = 0; wv < 8; ++wv) {
                const int c = waveCnt[wv];
                if (wv < wave) prefix += c;
                blockTotal += c;
            }
            if (m) posA[t] = runningBase + prefix + inWave;
            __syncthreads();
            if (tid == 0) runningBase += blockTotal;
            __syncthreads();
        }
        // After pass 0, runningBase == per-expert first-choice total == pos2 base.
    }
}

// ---------------------------------------------------------------------------
// 3) Dispatch: xe[e, pos, :] = x[t, :] for kept routes.
// ---------------------------------------------------------------------------
__global__ __launch_bounds__(256) void dispatch_kernel(
    const float* __restrict__ x,
    const int* __restrict__ idx1, const int* __restrict__ idx2,
    const int* __restrict__ pos1, const int* __restrict__ pos2,
    float* __restrict__ xe)
{
    const int t = blockIdx.x >> 1;
    const bool second = blockIdx.x & 1;
    const int p = second ? pos2[t] : pos1[t];
    if (p >= CAP) return;  // dropped token
    const int e = second ? idx2[t] : idx1[t];
    const int d4 = threadIdx.x * 4;
    const float4 v = *(const float4*)(x + (size_t)t * DIM + d4);
    *(float4*)(xe + ((size_t)e * CAP + p) * DIM + d4) = v;
}

// ---------------------------------------------------------------------------
// 4) Tiled fp32 WMMA GEMM:  C[M,N] = act(A[M,K] @ B[K,N] + bias[N])
//    Block tile 128x128, BK=16, 8 waves of 32x64, v_wmma_f32_16x16x4_f32.
//    Double-buffered LDS pipeline fed by GLOBAL_LOAD_ASYNC_TO_LDS (ASYNCcnt):
//    no VGPR staging -> no spills; B stored transposed [n][k] so every
//    fragment is one bank-clean ds_load_b64.
// ---------------------------------------------------------------------------
#define LPITCH 20   // LDS row pitch (floats): bank-clean b64 frags, 16B-aligned fills

template <int GELU>
__global__ __launch_bounds__(256) void moe_gemm_kernel(
    const float* __restrict__ A, const float* __restrict__ Bm,
    const float* __restrict__ bias, float* __restrict__ Cm,
    int M, int N, int K)
{
    __shared__ float As[2][128 * LPITCH];  // [m][k]
    __shared__ float Bs[2][128 * LPITCH];  // transposed: [n][k]

    const int tid  = threadIdx.x;
    const int lane = tid & 31;
    const int wave = tid >> 5;
    const int half = lane >> 4;      // 0: lanes 0-15, 1: lanes 16-31
    const int lr   = lane & 15;
    const int mWave = (wave & 3) * 32;
    const int nWave = (wave >> 2) * 64;
    const int tileM = blockIdx.y * 128;
    const int tileN = blockIdx.x * 128;

    v8f acc[2][4];
#pragma unroll
    for (int i = 0; i < 2; ++i)
#pragma unroll
        for (int j = 0; j < 4; ++j)
#pragma unroll
            for (int r = 0; r < 8; ++r) acc[i][j][r] = 0.0f;

#ifdef USE_ASYNC_LDS
    // Fire-and-forget copy of the (A,B) K-slab into LDS buffer `buf`.
    // A: 512 float4 units (b128 async). B: per-element b32 async so each lane
    // can write the transposed [n][k] destination; global side stays coalesced
    // (consecutive lanes -> consecutive n within one k row).
    auto issueAsync = [&](int k0, int buf) {
#pragma unroll
        for (int it = 0; it < 2; ++it) {
            const int idx  = tid + it * 256;         // 0..511
            const int arow = idx >> 2;
            const int acol = (idx & 3) << 2;
            __builtin_amdgcn_global_load_async_to_lds_b128(
                (gas128_t)(A + (size_t)(tileM + arow) * K + (k0 + acol)),
                (las128_t)&As[buf][arow * LPITCH + acol], 0, 0);
            const int bn  = idx & 127;
            const int bkq = idx >> 7;                // 0..3
#pragma unroll
            for (int i = 0; i < 4; ++i)
                __builtin_amdgcn_global_load_async_to_lds_b32(
                    (gas32_t)(Bm + (size_t)(k0 + bkq * 4 + i) * N + (tileN + bn)),
                    (las32_t)&Bs[buf][bn * LPITCH + bkq * 4 + i], 0, 0);
        }
    };
#else
    float4 aReg[2];
    float  bReg[2][4];
    auto loadGlobal = [&](int k0) {
#pragma unroll
        for (int it = 0; it < 2; ++it) {
            const int idx  = tid + it * 256;
            const int arow = idx >> 2;
            const int acol = (idx & 3) << 2;
            aReg[it] = *(const float4*)(A + (size_t)(tileM + arow) * K + (k0 + acol));
            const int bn  = idx & 127;
            const int bkq = idx >> 7;
#pragma unroll
            for (int i = 0; i < 4; ++i)
                bReg[it][i] = Bm[(size_t)(k0 + bkq * 4 + i) * N + (tileN + bn)];
        }
    };
    auto storeLds = [&](int buf) {
#pragma unroll
        for (int it = 0; it < 2; ++it) {
            const int idx  = tid + it * 256;
            const int arow = idx >> 2;
            const int acol = (idx & 3) << 2;
            *(float4*)(&As[buf][arow * LPITCH + acol]) = aReg[it];
            const int bn  = idx & 127;
            const int bkq = idx >> 7;
            float4 vb;
            vb.x = bReg[it][0]; vb.y = bReg[it][1];
            vb.z = bReg[it][2]; vb.w = bReg[it][3];
            *(float4*)(&Bs[buf][bn * LPITCH + bkq * 4]) = vb;
        }
    };
#endif

    auto compute = [&](int buf) {
#pragma unroll
        for (int ks = 0; ks < 4; ++ks) {
            // ISA frag layouts: lane-half selects K pair (K,K+1) vs (K+2,K+3)
            const int kb = ks * 4 + 2 * half;
            const v2f a0 = *(const v2f*)(&As[buf][(mWave + lr) * LPITCH + kb]);
            const v2f a1 = *(const v2f*)(&As[buf][(mWave + 16 + lr) * LPITCH + kb]);
#pragma unroll
            for (int ni = 0; ni < 4; ++ni) {
                const v2f b = *(const v2f*)(&Bs[buf][(nWave + ni * 16 + lr) * LPITCH + kb]);
                acc[0][ni] = __builtin_amdgcn_wmma_f32_16x16x4_f32(
                    false, a0, false, b, (short)0, acc[0][ni], false, false);
                acc[1][ni] = __builtin_amdgcn_wmma_f32_16x16x4_f32(
                    false, a1, false, b, (short)0, acc[1][ni], false, false);
            }
        }
    };

#ifdef USE_ASYNC_LDS
    issueAsync(0, 0);
    wait_async0();
    __syncthreads();
    int buf = 0;
    for (int k0 = 0; k0 < K; k0 += 16) {
        const bool hasNext = (k0 + 16) < K;
        if (hasNext) issueAsync(k0 + 16, buf ^ 1);   // overlaps WMMA below
        compute(buf);
        wait_async0();
        __syncthreads();
        buf ^= 1;
    }
#else
    loadGlobal(0);
    storeLds(0);
    __syncthreads();
    int buf = 0;
    for (int k0 = 0; k0 < K; k0 += 16) {
        const bool hasNext = (k0 + 16) < K;
        if (hasNext) loadGlobal(k0 + 16);
        compute(buf);
        if (hasNext) storeLds(buf ^ 1);
        __syncthreads();
        buf ^= 1;
    }
#endif

    // Epilogue: bias + (optional) exact GELU.  C/D layout: VGPR r -> rows r, r+8.
#pragma unroll
    for (int mi = 0; mi < 2; ++mi)
#pragma unroll
        for (int ni = 0; ni < 4; ++ni) {
            const int nG = tileN + nWave + ni * 16 + lr;
            const float bv = bias[nG];
#pragma unroll
            for (int r = 0; r < 8; ++r) {
                const int mG = tileM + mWave + mi * 16 + half * 8 + r;
                float v = acc[mi][ni][r] + bv;
                if (GELU) v = 0.5f * v * (1.0f + erff(v * 0.70710678118654752f));
                Cm[(size_t)mG * N + nG] = v;
            }
        }
}

// ---------------------------------------------------------------------------
// 5) Combine: out[t] = w1*ye[idx1,pos1] + w2*ye[idx2,pos2] (0 if dropped).
// ---------------------------------------------------------------------------
__global__ __launch_bounds__(256) void combine_kernel(
    const float* __restrict__ ye,
    const int* __restrict__ idx1, const int* __restrict__ idx2,
    const int* __restrict__ pos1, const int* __restrict__ pos2,
    const float* __restrict__ w1, const float* __restrict__ w2,
    float* __restrict__ out)
{
    const int t = blockIdx.x;
    const int d4 = threadIdx.x * 4;
    const int p1 = pos1[t], p2 = pos2[t];
    const float a = (p1 < CAP) ? w1[t] : 0.0f;
    const float b = (p2 < CAP) ? w2[t] : 0.0f;
    const int pc1 = p1 < CAP ? p1 : (CAP - 1);
    const int pc2 = p2 < CAP ? p2 : (CAP - 1);
    const float4 y1 = *(const float4*)(ye + ((size_t)idx1[t] * CAP + pc1) * DIM + d4);
    const float4 y2 = *(const float4*)(ye + ((size_t)idx2[t] * CAP + pc2) * DIM + d4);
    float4 o;
    o.x = a * y1.x + b * y2.x;
    o.y = a * y1.y + b * y2.y;
    o.z = a * y1.z + b * y2.z;
    o.w = a * y1.w + b * y2.w;
    *(float4*)(out + (size_t)t * DIM + d4) = o;
}

// ---------------------------------------------------------------------------
extern "C" void kernel_launch(void* const* d_in, const int* in_sizes, int n_in,
                              void* d_out, int out_size, void* d_ws, size_t ws_size,
                              hipStream_t stream)
{
    (void)in_sizes; (void)n_in; (void)out_size; (void)ws_size;
    const float* x  = (const float*)d_in[0];
    const float* Wg = (const float*)d_in[1];
    const float* W1 = (const float*)d_in[2];
    const float* b1 = (const float*)d_in[3];
    const float* W2 = (const float*)d_in[4];
    const float* b2 = (const float*)d_in[5];
    float* out = (float*)d_out;

    char* ws = (char*)d_ws;
    size_t off = 0;
    auto take = [&](size_t bytes) -> char* {
        char* p = ws + off;
        off = (off + bytes + 255) & ~(size_t)255;
        return p;
    };
    int*   idx1 = (int*)take((size_t)TOK * 4);
    int*   idx2 = (int*)take((size_t)TOK * 4);
    int*   pos1 = (int*)take((size_t)TOK * 4);
    int*   pos2 = (int*)take((size_t)TOK * 4);
    float* w1   = (float*)take((size_t)TOK * 4);
    float* w2   = (float*)take((size_t)TOK * 4);
    float* xe   = (float*)take((size_t)NEXP * CAP * DIM * 4);  // 64 MiB
    float* ye   = (float*)take((size_t)NEXP * CAP * DIM * 4);  // 64 MiB
    float* h    = (float*)take((size_t)CAP * FF * 4);          // 32 MiB, reused per expert

    gate_kernel<<<TOK / 8, 256, 0, stream>>>(x, Wg, idx1, idx2, w1, w2);
    scan_kernel<<<NEXP, 256, 0, stream>>>(idx1, idx2, pos1, pos2);
    dispatch_kernel<<<2 * TOK, 256, 0, stream>>>(x, idx1, idx2, pos1, pos2, xe);

    for (int e = 0; e < NEXP; ++e) {
        const float* Ae = xe + (size_t)e * CAP * DIM;
        moe_gemm_kernel<1><<<dim3(FF / 128, CAP / 128, 1), 256, 0, stream>>>(
            Ae, W1 + (size_t)e * DIM * FF, b1 + (size_t)e * FF, h, CAP, FF, DIM);
        moe_gemm_kernel<0><<<dim3(DIM / 128, CAP / 128, 1), 256, 0, stream>>>(
            h, W2 + (size_t)e * FF * DIM, b2 + (size_t)e * DIM,
            ye + (size_t)e * CAP * DIM, CAP, DIM, FF);
    }

    combine_kernel<<<TOK, 256, 0, stream>>>(ye, idx1, idx2, pos1, pos2, w1, w2, out);
}